// TVNet_25915832664507
// MI455X (gfx1250) — compile-verified
//
#include <hip/hip_runtime.h>
#include <hip/hip_bf16.h>
#include <stdint.h>

// ---------------------------------------------------------------------------
// TVNet (TV-L1 optical flow), B=8, H=W=512, fp32.
// Memory-bound; entire working set (~96MB) fits MI455X's 192MB L2.
// CDNA5 paths used: global_load_async_to_lds_b32 + s_wait_asynccnt (LDS tile
// staging for the stencil), global_prefetch_b8.
// ---------------------------------------------------------------------------

#define WID 512
#define HEI 512
#define HW  (WID * HEI)

#define EPS_F 1e-12f
#define THETA_F 0.3f
static constexpr float L_T  = (float)(0.15 * 0.3);   // LBDA * THETA
static constexpr float TAUT = (float)(0.25 / 0.3);   // TAU / THETA

#define BX 32
#define BY 8

// Detect CDNA5 async global->LDS builtins (gfx1250)
#if defined(__has_builtin)
#if __has_builtin(__builtin_amdgcn_global_load_async_to_lds_b32)
#define HAVE_ASYNC_LDS 1
#endif
#endif

#ifdef HAVE_ASYNC_LDS
typedef __attribute__((address_space(1))) int GInt;   // global (AS1)
typedef __attribute__((address_space(3))) int LInt;   // LDS (AS3)
__device__ __forceinline__ void async_wait_all() {
#if __has_builtin(__builtin_amdgcn_s_wait_asynccnt)
  __builtin_amdgcn_s_wait_asynccnt(0);
#else
  asm volatile("s_wait_asynccnt 0" ::: "memory");
#endif
}
#endif

// ---------------------------------------------------------------------------
// k_init: working u1/u2 <- inputs; zero p11..p22 and rho.
// ---------------------------------------------------------------------------
__global__ void k_init(const float* __restrict__ u1in, const float* __restrict__ u2in,
                       float* __restrict__ u1, float* __restrict__ u2,
                       float* __restrict__ rho,
                       float* __restrict__ p11, float* __restrict__ p12,
                       float* __restrict__ p21, float* __restrict__ p22, int n) {
  int i = blockIdx.x * blockDim.x + threadIdx.x;
  if (i >= n) return;
  u1[i] = u1in[i];
  u2[i] = u2in[i];
  rho[i] = 0.0f;
  p11[i] = 0.0f; p12[i] = 0.0f; p21[i] = 0.0f; p22[i] = 0.0f;
}

// ---------------------------------------------------------------------------
// k_cgrad: centered gradient of x2 (with replicated-border one-sided stencil,
// exactly the reference's concatenate form via clamped indices).
// ---------------------------------------------------------------------------
__global__ void k_cgrad(const float* __restrict__ x2,
                        float* __restrict__ d2x, float* __restrict__ d2y, int n) {
  int i = blockIdx.x * blockDim.x + threadIdx.x;
  if (i >= n) return;
  int c = i & (WID - 1);
  int r = (i >> 9) & (HEI - 1);
  int base = i & ~(HW - 1);
  int cl = max(c - 1, 0), cr = min(c + 1, WID - 1);
  int rl = max(r - 1, 0), rr = min(r + 1, HEI - 1);
  d2x[i] = 0.5f * (x2[base + r * WID + cr] - x2[base + r * WID + cl]);
  d2y[i] = 0.5f * (x2[base + rr * WID + c] - x2[base + rl * WID + c]);
}

// ---------------------------------------------------------------------------
// k_warp: bilinear warp of x2, d2x, d2y by (u1,u2); emit dxw, dyw and
// rho_c = x2w - dxw*u1 - dyw*u2 - x1  (x2w folded, never stored).
// Out-of-range samples contribute 0 (validity per-corner like the reference).
// ---------------------------------------------------------------------------
__global__ void k_warp(const float* __restrict__ x1, const float* __restrict__ x2,
                       const float* __restrict__ d2x, const float* __restrict__ d2y,
                       const float* __restrict__ u1, const float* __restrict__ u2,
                       float* __restrict__ dxw, float* __restrict__ dyw,
                       float* __restrict__ rho_c, int n) {
  int i = blockIdx.x * blockDim.x + threadIdx.x;
  if (i >= n) return;
  int c = i & (WID - 1);
  int r = (i >> 9) & (HEI - 1);
  int base = i & ~(HW - 1);

  float u1v = u1[i], u2v = u2[i];
  float gx = (float)c + u1v;
  float gy = (float)r + u2v;
  // Clamp to a safe range: anything below -1 or above W is invalid for both
  // corners anyway, so clamping preserves the reference's validity semantics
  // while keeping the int conversion well-defined.
  gx = fminf(fmaxf(gx, -8.0f), (float)(WID + 8));
  gy = fminf(fmaxf(gy, -8.0f), (float)(HEI + 8));
  float x0f = floorf(gx), y0f = floorf(gy);
  float wx = gx - x0f, wy = gy - y0f;
  int ix0 = (int)x0f, iy0 = (int)y0f;
  int ix1 = ix0 + 1, iy1 = iy0 + 1;
  bool vx0 = (ix0 >= 0) && (ix0 < WID);
  bool vx1 = (ix1 >= 0) && (ix1 < WID);
  bool vy0 = (iy0 >= 0) && (iy0 < HEI);
  bool vy1 = (iy1 >= 0) && (iy1 < HEI);
  int cx0 = min(max(ix0, 0), WID - 1), cx1 = min(max(ix1, 0), WID - 1);
  int cy0 = min(max(iy0, 0), HEI - 1), cy1 = min(max(iy1, 0), HEI - 1);
  int o00 = base + cy0 * WID + cx0;
  int o01 = base + cy0 * WID + cx1;
  int o10 = base + cy1 * WID + cx0;
  int o11 = base + cy1 * WID + cx1;
  float w00 = (1.0f - wx) * (1.0f - wy);
  float w01 = wx * (1.0f - wy);
  float w10 = (1.0f - wx) * wy;
  float w11 = wx * wy;
  bool v00 = vx0 && vy0, v01 = vx1 && vy0, v10 = vx0 && vy1, v11 = vx1 && vy1;

#define SAMP(im) (w00 * (v00 ? im[o00] : 0.0f) + w01 * (v01 ? im[o01] : 0.0f) + \
                  w10 * (v10 ? im[o10] : 0.0f) + w11 * (v11 ? im[o11] : 0.0f))
  float x2w = SAMP(x2);
  float dxv = SAMP(d2x);
  float dyv = SAMP(d2y);
#undef SAMP

  dxw[i] = dxv;
  dyw[i] = dyv;
  rho_c[i] = x2w - dxv * u1v - dyv * u2v - x1[i];
}

// ---------------------------------------------------------------------------
// k_update_u: thresholding step + u <- v + THETA * div(p).  In-place on u
// (each pixel only reads its own u).  grad recomputed from dxw/dyw (saves a
// full array of L2 traffic).  rho written out every call; the last call
// leaves the reference's final rho in d_out.
// ---------------------------------------------------------------------------
__global__ void k_update_u(const float* __restrict__ dxw, const float* __restrict__ dyw,
                           const float* __restrict__ rho_c,
                           const float* __restrict__ p11, const float* __restrict__ p12,
                           const float* __restrict__ p21, const float* __restrict__ p22,
                           float* __restrict__ u1, float* __restrict__ u2,
                           float* __restrict__ rho_out, int n) {
  int i = blockIdx.x * blockDim.x + threadIdx.x;
  if (i >= n) return;

  // Warm L2/L0 for the p arrays we'll read below (global_prefetch_b8).
  __builtin_prefetch(p11 + i, 0, 1);
  __builtin_prefetch(p12 + i, 0, 1);
  __builtin_prefetch(p21 + i, 0, 1);
  __builtin_prefetch(p22 + i, 0, 1);

  int c = i & (WID - 1);
  int r = (i >> 9) & (HEI - 1);

  float dxv = dxw[i], dyv = dyw[i], rcv = rho_c[i];
  float u1v = u1[i], u2v = u2[i];
  float grad = dxv * dxv + dyv * dyv + EPS_F;
  float rho = rcv + dxv * u1v + dyv * u2v + EPS_F;

  float lt_g = L_T * grad;
  float scale;
  if (rho < -lt_g)        scale = L_T;
  else if (rho > lt_g)    scale = -L_T;
  else if (grad > EPS_F)  scale = -rho / grad;
  else                    scale = 0.0f;

  float v1 = u1v + scale * dxv;
  float v2 = u2v + scale * dyv;

  // forward divergence: dx[0]=p[0]; dx[c]=p[c]-p[c-1]; dx[W-1]=-p[W-2]
  float p11v = p11[i], p12v = p12[i], p21v = p21[i], p22v = p22[i];
  float dpx1 = (c == 0) ? p11v : ((c == WID - 1) ? -p11[i - 1] : p11v - p11[i - 1]);
  float dpy1 = (r == 0) ? p12v : ((r == HEI - 1) ? -p12[i - WID] : p12v - p12[i - WID]);
  float dpx2 = (c == 0) ? p21v : ((c == WID - 1) ? -p21[i - 1] : p21v - p21[i - 1]);
  float dpy2 = (r == 0) ? p22v : ((r == HEI - 1) ? -p22[i - WID] : p22v - p22[i - WID]);

  u1[i] = v1 + THETA_F * (dpx1 + dpy1);
  u2[i] = v2 + THETA_F * (dpx2 + dpy2);
  rho_out[i] = rho;
}

// ---------------------------------------------------------------------------
// k_update_p: forward gradient of u (right/down neighbors) via an LDS-staged
// (BX+1)x(BY+1) tile per array, loaded with CDNA5 async global->LDS copies,
// then p <- (p + taut*grad(u)) / (1 + taut*|grad(u)|).  In-place on p.
// ---------------------------------------------------------------------------
__global__ void k_update_p(const float* __restrict__ u1, const float* __restrict__ u2,
                           float* __restrict__ p11, float* __restrict__ p12,
                           float* __restrict__ p21, float* __restrict__ p22) {
  __shared__ float su1[(BY + 1) * (BX + 1)];
  __shared__ float su2[(BY + 1) * (BX + 1)];

  const int tx = threadIdx.x, ty = threadIdx.y;
  const int c0 = blockIdx.x * BX, r0 = blockIdx.y * BY;
  const int base = blockIdx.z * HW;
  const int tid = ty * BX + tx;
  const int nElem = (BY + 1) * (BX + 1);  // 297

  for (int t = tid; t < nElem; t += BX * BY) {
    int ly = t / (BX + 1);
    int lx = t - ly * (BX + 1);
    int gxc = min(c0 + lx, WID - 1);   // clamped halo; never consumed when OOB
    int gyc = min(r0 + ly, HEI - 1);
    int gi = base + gyc * WID + gxc;
#ifdef HAVE_ASYNC_LDS
    __builtin_amdgcn_global_load_async_to_lds_b32((GInt*)(u1 + gi), (LInt*)(su1 + t), 0, 0);
    __builtin_amdgcn_global_load_async_to_lds_b32((GInt*)(u2 + gi), (LInt*)(su2 + t), 0, 0);
#else
    su1[t] = u1[gi];
    su2[t] = u2[gi];
#endif
  }
#ifdef HAVE_ASYNC_LDS
  async_wait_all();
#endif
  __syncthreads();

  const int c = c0 + tx, r = r0 + ty;
  const int li = ty * (BX + 1) + tx;

  float u1v = su1[li];
  float u2v = su2[li];
  float u1r = su1[li + 1],        u1d = su1[li + (BX + 1)];
  float u2r = su2[li + 1],        u2d = su2[li + (BX + 1)];
  float u1x = (c == WID - 1) ? 0.0f : u1r - u1v;
  float u1y = (r == HEI - 1) ? 0.0f : u1d - u1v;
  float u2x = (c == WID - 1) ? 0.0f : u2r - u2v;
  float u2y = (r == HEI - 1) ? 0.0f : u2d - u2v;

  float n1 = 1.0f + TAUT * sqrtf(u1x * u1x + u1y * u1y + EPS_F);
  float n2 = 1.0f + TAUT * sqrtf(u2x * u2x + u2y * u2y + EPS_F);

  int i = base + r * WID + c;
  p11[i] = (p11[i] + TAUT * u1x) / n1;
  p12[i] = (p12[i] + TAUT * u1y) / n1;
  p21[i] = (p21[i] + TAUT * u2x) / n2;
  p22[i] = (p22[i] + TAUT * u2y) / n2;
}

// ---------------------------------------------------------------------------
// Host side
// ---------------------------------------------------------------------------
extern "C" void kernel_launch(void* const* d_in, const int* in_sizes, int n_in,
                              void* d_out, int out_size, void* d_ws, size_t ws_size,
                              hipStream_t stream) {
  (void)n_in; (void)out_size; (void)ws_size;
  const float* x1   = (const float*)d_in[0];
  const float* x2   = (const float*)d_in[1];
  const float* u1in = (const float*)d_in[2];
  const float* u2in = (const float*)d_in[3];

  const int n = in_sizes[0];           // B*H*W = 8*512*512
  const int Bn = n / HW;

  float* out_u1  = (float*)d_out;
  float* out_u2  = out_u1 + (size_t)n;
  float* out_rho = out_u1 + 2 * (size_t)n;

  float* ws = (float*)d_ws;            // 9 arrays * n floats = 72 MB
  const size_t N = (size_t)n;
  float* d2x   = ws + 0 * N;
  float* d2y   = ws + 1 * N;
  float* dxw   = ws + 2 * N;
  float* dyw   = ws + 3 * N;
  float* rho_c = ws + 4 * N;
  float* p11   = ws + 5 * N;
  float* p12   = ws + 6 * N;
  float* p21   = ws + 7 * N;
  float* p22   = ws + 8 * N;

  const int threads = 256;
  const int blocks = (n + threads - 1) / threads;
  dim3 pgrid(WID / BX, HEI / BY, Bn);
  dim3 pblk(BX, BY, 1);

  k_init<<<blocks, threads, 0, stream>>>(u1in, u2in, out_u1, out_u2, out_rho,
                                         p11, p12, p21, p22, n);
  k_cgrad<<<blocks, threads, 0, stream>>>(x2, d2x, d2y, n);

  for (int w = 0; w < 5; ++w) {          // N_WARPS
    k_warp<<<blocks, threads, 0, stream>>>(x1, x2, d2x, d2y, out_u1, out_u2,
                                           dxw, dyw, rho_c, n);
    for (int it = 0; it < 5; ++it) {     // N_ITERS
      k_update_u<<<blocks, threads, 0, stream>>>(dxw, dyw, rho_c,
                                                 p11, p12, p21, p22,
                                                 out_u1, out_u2, out_rho, n);
      k_update_p<<<pgrid, pblk, 0, stream>>>(out_u1, out_u2, p11, p12, p21, p22);
    }
  }
}